// GCNBasicModel_45200235823717
// MI455X (gfx1250) — compile-verified
//
#include <hip/hip_runtime.h>
#include <math.h>

typedef float v2f __attribute__((ext_vector_type(2)));
typedef float v8f __attribute__((ext_vector_type(8)));

// ---------------------------------------------------------------------------
// Utility: zero a buffer (float4 granularity; counts are multiples of 4)
// ---------------------------------------------------------------------------
__global__ void zero4_kernel(float4* __restrict__ p, long n4) {
    long i = (long)blockIdx.x * blockDim.x + threadIdx.x;
    if (i < n4) p[i] = make_float4(0.f, 0.f, 0.f, 0.f);
}

// ---------------------------------------------------------------------------
// Degree accumulation: deg[dst[e]] += 1
// ---------------------------------------------------------------------------
__global__ void deg_kernel(const int* __restrict__ dst, float* __restrict__ deg, int E) {
    int e = blockIdx.x * blockDim.x + threadIdx.x;
    if (e < E) atomicAdd(&deg[dst[e]], 1.0f);
}

// dinv[i] = rsqrt(deg[i] + 1)   (+1 = self loop)
__global__ void dinv_kernel(float* __restrict__ deg, int N) {
    int i = blockIdx.x * blockDim.x + threadIdx.x;
    if (i < N) deg[i] = rsqrtf(deg[i] + 1.0f);
}

// ---------------------------------------------------------------------------
// GEMM: C[M,128] = A[M,128] @ W[128,128]  using V_WMMA_F32_16X16X4_F32.
// 256 threads/block = 8 waves; block covers 128 rows, each wave a 16x128 strip.
//
// W staged in LDS with even/odd K rows INTERLEAVED:
//   sW[(r>>1)*256 + c*2 + (r&1)] = W[r][c]
// so a B fragment (rows k+kh, k+kh+1 at column c) is ONE contiguous 8-byte
// LDS load -> ds_load_b64 into an even-aligned VGPR pair, no re-pairing movs.
//
// A-frag (16x4 f32):  lane<16 -> M=lane,   K = k..k+1 in v0,v1
//                     lane>=16 -> M=lane-16, K = k+2..k+3
// B-frag (4x16 f32):  v0: row k (lanes 0-15) / row k+2 (lanes 16-31); v1: k+1/k+3
// C (16x16 f32):      vgpr v: lanes 0-15 -> M=v, N=lane; lanes 16-31 -> M=v+8
// ---------------------------------------------------------------------------
__global__ __launch_bounds__(256)
void gemm_x128_kernel(const float* __restrict__ A, const float* __restrict__ W,
                      float* __restrict__ C, int M) {
    __shared__ float sW[128 * 128];
    for (int idx = threadIdx.x; idx < 128 * 128; idx += 256) {
        const int r = idx >> 7;
        const int c = idx & 127;
        sW[((r >> 1) << 8) + (c << 1) + (r & 1)] = W[idx];
    }
    __syncthreads();

    const int wave = threadIdx.x >> 5;
    const int lane = threadIdx.x & 31;
    const int r0 = blockIdx.x * 128 + wave * 16;
    if (r0 >= M) return;                       // wave-uniform; M % 16 == 0

    const int ml = lane & 15;                  // M within tile / N within tile
    const int kh = (lane >> 4) * 2;            // K sub-offset: 0 or 2
    const int rh = (lane >> 4) * 8;            // C row half offset

    v8f acc[8];
    const v8f z8 = {0.f, 0.f, 0.f, 0.f, 0.f, 0.f, 0.f, 0.f};
#pragma unroll
    for (int n = 0; n < 8; ++n) acc[n] = z8;

    const float* Arow = A + (size_t)(r0 + ml) * 128;

#pragma unroll 4
    for (int k = 0; k < 128; k += 4) {
        v2f a;
        a.x = Arow[k + kh];
        a.y = Arow[k + kh + 1];
        // pair-row (k+kh)/2, 128 v2f entries per pair-row
        const v2f* wp = ((const v2f*)sW) + (((k + kh) >> 1) << 7) + ml;
#pragma unroll
        for (int n = 0; n < 8; ++n) {
            v2f b = wp[n * 16];
            acc[n] = __builtin_amdgcn_wmma_f32_16x16x4_f32(
                false, a, false, b, (short)0, acc[n], false, false);
        }
    }

#pragma unroll
    for (int n = 0; n < 8; ++n) {
        float* Cp = C + (size_t)(r0 + rh) * 128 + n * 16 + ml;
#pragma unroll
        for (int v = 0; v < 8; ++v) Cp[(size_t)v * 128] = acc[n][v];
    }
}

// ---------------------------------------------------------------------------
// FC GEMM: logits[M,40] = A[M,128] @ Wfc[128,40]
// N padded to 48 (3 tiles) with zero-fill done once in LDS, so the WMMA inner
// loop has no guards. Same interleaved even/odd K-row LDS layout (48 cols).
// ---------------------------------------------------------------------------
__global__ __launch_bounds__(256)
void gemm_fc_kernel(const float* __restrict__ A, const float* __restrict__ W,
                    float* __restrict__ C, int M) {
    __shared__ float sW[64 * 48 * 2];          // 64 pair-rows x 48 cols x 2
    for (int idx = threadIdx.x; idx < 64 * 48 * 2; idx += 256) sW[idx] = 0.f;
    __syncthreads();
    for (int idx = threadIdx.x; idx < 128 * 40; idx += 256) {
        const int r = idx / 40;
        const int c = idx - r * 40;
        sW[(r >> 1) * 96 + (c << 1) + (r & 1)] = W[idx];
    }
    __syncthreads();

    const int wave = threadIdx.x >> 5;
    const int lane = threadIdx.x & 31;
    const int r0 = blockIdx.x * 128 + wave * 16;
    if (r0 >= M) return;

    const int ml = lane & 15;
    const int kh = (lane >> 4) * 2;
    const int rh = (lane >> 4) * 8;

    v8f acc[3];
    const v8f z8 = {0.f, 0.f, 0.f, 0.f, 0.f, 0.f, 0.f, 0.f};
#pragma unroll
    for (int n = 0; n < 3; ++n) acc[n] = z8;

    const float* Arow = A + (size_t)(r0 + ml) * 128;

#pragma unroll 4
    for (int k = 0; k < 128; k += 4) {
        v2f a;
        a.x = Arow[k + kh];
        a.y = Arow[k + kh + 1];
        const v2f* wp = ((const v2f*)sW) + ((k + kh) >> 1) * 48 + ml;
#pragma unroll
        for (int n = 0; n < 3; ++n) {
            v2f b = wp[n * 16];
            acc[n] = __builtin_amdgcn_wmma_f32_16x16x4_f32(
                false, a, false, b, (short)0, acc[n], false, false);
        }
    }

#pragma unroll
    for (int n = 0; n < 3; ++n) {
        const int c = n * 16 + ml;
        if (c < 40) {
            float* Cp = C + (size_t)(r0 + rh) * 40 + c;
#pragma unroll
            for (int v = 0; v < 8; ++v) Cp[(size_t)v * 40] = acc[n][v];
        }
    }
}

// ---------------------------------------------------------------------------
// Edge scatter: one wave per edge.
// agg[dst] += dinv[src]*dinv[dst] * h[src]   (128 f32 features = 1 float4/lane)
// ---------------------------------------------------------------------------
__global__ __launch_bounds__(256)
void edge_scatter_kernel(const int* __restrict__ src, const int* __restrict__ dst,
                         const float* __restrict__ dinv, const float* __restrict__ h,
                         float* __restrict__ agg, int E) {
    int e = blockIdx.x * 8 + (threadIdx.x >> 5);
    if (e >= E) return;
    const int lane = threadIdx.x & 31;
    const int s = src[e];
    const int d = dst[e];
    const float nrm = dinv[s] * dinv[d];
    const float4 v = ((const float4*)(h + (size_t)s * 128))[lane];
    float* o = agg + (size_t)d * 128 + lane * 4;
    atomicAdd(o + 0, nrm * v.x);
    atomicAdd(o + 1, nrm * v.y);
    atomicAdd(o + 2, nrm * v.z);
    atomicAdd(o + 3, nrm * v.w);
}

// ---------------------------------------------------------------------------
// Fused self-loop + bias + ReLU:
//   h[i,f] = relu(agg[i,f] + dinv[i]^2 * h[i,f] + b[f])   (float4 granularity)
// ---------------------------------------------------------------------------
__global__ void fuse_relu_kernel(const float* __restrict__ agg, float* __restrict__ h,
                                 const float* __restrict__ dinv, const float* __restrict__ bias,
                                 int N) {
    int i = blockIdx.x * blockDim.x + threadIdx.x;   // over N*32 float4 chunks
    if (i >= N * 32) return;
    const int node = i >> 5;
    const int c4 = i & 31;
    const float dv = dinv[node];
    const float dd = dv * dv;
    const float4 a = ((const float4*)agg)[i];
    const float4 hv = ((const float4*)h)[i];
    const float4 bv = ((const float4*)bias)[c4];
    float4 r;
    r.x = fmaxf(a.x + dd * hv.x + bv.x, 0.f);
    r.y = fmaxf(a.y + dd * hv.y + bv.y, 0.f);
    r.z = fmaxf(a.z + dd * hv.z + bv.z, 0.f);
    r.w = fmaxf(a.w + dd * hv.w + bv.w, 0.f);
    ((float4*)h)[i] = r;
}

// ---------------------------------------------------------------------------
// Row-wise log_softmax over 40 logits (bias added here), in place on d_out.
// ---------------------------------------------------------------------------
__global__ void logsoftmax_kernel(float* __restrict__ io, const float* __restrict__ bfc, int N) {
    int r = blockIdx.x * blockDim.x + threadIdx.x;
    if (r >= N) return;
    float* row = io + (size_t)r * 40;
    float m = -INFINITY;
    for (int j = 0; j < 40; ++j) m = fmaxf(m, row[j] + bfc[j]);
    float s = 0.f;
    for (int j = 0; j < 40; ++j) s += expf(row[j] + bfc[j] - m);
    const float ls = logf(s);
    for (int j = 0; j < 40; ++j) row[j] = row[j] + bfc[j] - m - ls;
}

// ---------------------------------------------------------------------------
// Launcher
// ---------------------------------------------------------------------------
extern "C" void kernel_launch(void* const* d_in, const int* in_sizes, int n_in,
                              void* d_out, int out_size, void* d_ws, size_t ws_size,
                              hipStream_t stream) {
    const float* x   = (const float*)d_in[0];
    const int*  edge = (const int*)d_in[1];
    const float* W1  = (const float*)d_in[2];
    const float* b1  = (const float*)d_in[3];
    const float* W2  = (const float*)d_in[4];
    const float* b2  = (const float*)d_in[5];
    const float* Wfc = (const float*)d_in[6];
    const float* bfc = (const float*)d_in[7];

    const int N = in_sizes[0] / 128;   // 100000
    const int E = in_sizes[1] / 2;     // 1600000
    const int* srcI = edge;            // edge_index[0]
    const int* dstI = edge + E;        // edge_index[1]

    float* dinv = (float*)d_ws;                                  // N floats
    float* bufA = (float*)((char*)d_ws + (size_t)(512 << 10));   // N*128 floats
    float* bufB = bufA + (size_t)N * 128;                        // N*128 floats
    float* out  = (float*)d_out;                                 // N*40 floats

    const long n4feat = (long)N * 32;          // N*128/4 float4 chunks
    const int  TB = 256;

    // degrees -> dinv
    zero4_kernel<<<(int)((N / 4 + TB - 1) / TB), TB, 0, stream>>>((float4*)dinv, N / 4);
    deg_kernel<<<(E + TB - 1) / TB, TB, 0, stream>>>(dstI, dinv, E);
    dinv_kernel<<<(N + TB - 1) / TB, TB, 0, stream>>>(dinv, N);

    // ---- layer 1: h = relu(Ahat @ (x @ W1) + b1) ----
    gemm_x128_kernel<<<(N + 127) / 128, TB, 0, stream>>>(x, W1, bufB, N);
    zero4_kernel<<<(int)((n4feat + TB - 1) / TB), TB, 0, stream>>>((float4*)bufA, n4feat);
    edge_scatter_kernel<<<(E + 7) / 8, TB, 0, stream>>>(srcI, dstI, dinv, bufB, bufA, E);
    fuse_relu_kernel<<<(int)(((long)N * 32 + TB - 1) / TB), TB, 0, stream>>>(bufA, bufB, dinv, b1, N);

    // ---- layer 2 ----
    gemm_x128_kernel<<<(N + 127) / 128, TB, 0, stream>>>(bufB, W2, bufA, N);
    zero4_kernel<<<(int)((n4feat + TB - 1) / TB), TB, 0, stream>>>((float4*)bufB, n4feat);
    edge_scatter_kernel<<<(E + 7) / 8, TB, 0, stream>>>(srcI, dstI, dinv, bufA, bufB, E);
    fuse_relu_kernel<<<(int)(((long)N * 32 + TB - 1) / TB), TB, 0, stream>>>(bufB, bufA, dinv, b2, N);

    // ---- FC head + log_softmax ----
    gemm_fc_kernel<<<(N + 127) / 128, TB, 0, stream>>>(bufA, Wfc, out, N);
    logsoftmax_kernel<<<(N + TB - 1) / TB, TB, 0, stream>>>(out, bfc, N);
}